// _DCN_Module_395136991332
// MI455X (gfx1250) — compile-verified
//
#include <hip/hip_runtime.h>

#define N_PTS 65536
#define K_CL  1024
#define D_DIM 512

typedef __attribute__((ext_vector_type(2))) float v2f;
typedef __attribute__((ext_vector_type(8))) float v8f;

// ---------------------------------------------------------------------------
// Kernel: zero scratch (added[K*D], n_add[K] (as bits), loss accumulator)
// ---------------------------------------------------------------------------
__global__ __launch_bounds__(256) void zero_kernel(float* __restrict__ p, int n) {
    int i = blockIdx.x * 256 + threadIdx.x;
    if (i < n) p[i] = 0.0f;
}

// ---------------------------------------------------------------------------
// Kernel: c2[k] = sum_d centers[k][d]^2   (one wave32 per row)
// ---------------------------------------------------------------------------
__global__ __launch_bounds__(256) void c2_kernel(const float* __restrict__ C,
                                                 float* __restrict__ c2) {
    const int lane = threadIdx.x & 31;
    const int w    = threadIdx.x >> 5;
    const int k    = blockIdx.x * 8 + w;
    const float* row = C + (size_t)k * D_DIM;
    float s = 0.0f;
#pragma unroll
    for (int i = 0; i < D_DIM / 128; ++i) {            // b128 per lane
        const float4 v = ((const float4*)row)[lane + 32 * i];
        s += v.x * v.x + v.y * v.y + v.z * v.z + v.w * v.w;
    }
#pragma unroll
    for (int off = 16; off > 0; off >>= 1) s += __shfl_xor(s, off, 32);
    if (lane == 0) c2[k] = s;
}

// ---------------------------------------------------------------------------
// Main kernel: fp32 WMMA GEMM (E @ C^T) fused with per-row min/argmin.
// Block = 256 threads = 8 waves; each wave owns 32 rows of E (two 16-row
// A fragments) and sweeps all K=1024 clusters in column-groups of 64
// (4 column tiles -> 8 accumulators, A fragments reused 4x each).
//
// WMMA f32 16x16x4 fragment layout (wave32):
//   A (16x4):  lane l<16 : row=l,    a={A[r][d+0],A[r][d+1]}
//              lane l>=16: row=l-16, a={A[r][d+2],A[r][d+3]}
//   B (4x16):  lane l<16 : col=l,    b={B[d+0][c],B[d+1][c]}  (B = C^T =>
//              lane l>=16: col=l-16, b={B[d+2][c],B[d+3][c]}   contiguous in C)
//   D (16x16): lane l, vgpr j -> (row = j + 8*(l>>4), col = l&15)
// ---------------------------------------------------------------------------
__global__ __launch_bounds__(256) void assign_kernel(
    const float* __restrict__ E, const float* __restrict__ C,
    const float* __restrict__ c2, float* __restrict__ s_out,
    float* __restrict__ m_out, int* __restrict__ s_int)
{
    const int lane = threadIdx.x & 31;
    const int w    = threadIdx.x >> 5;
    const int half = lane >> 4;       // 0 or 1
    const int l16  = lane & 15;
    const int n0   = blockIdx.x * 256 + w * 32;

    const float* A0 = E + (size_t)(n0 + l16) * D_DIM + 2 * half;
    const float* A1 = A0 + (size_t)16 * D_DIM;

    float vmin[2][8];
    int   amin[2][8];
#pragma unroll
    for (int u = 0; u < 2; ++u)
#pragma unroll
        for (int j = 0; j < 8; ++j) { vmin[u][j] = 3.4e38f; amin[u][j] = 0; }

    for (int cg = 0; cg < K_CL; cg += 64) {
        const float* Bp = C + (size_t)(cg + l16) * D_DIM + 2 * half;

        // Speculative prefetch of the next column group's center rows.
        if (cg + 64 < K_CL) {
            const float* Pn = C + (size_t)(cg + 64 + l16) * D_DIM;
            __builtin_prefetch(Pn, 0, 3);
            __builtin_prefetch(Pn + (size_t)16 * D_DIM, 0, 3);
            __builtin_prefetch(Pn + (size_t)32 * D_DIM, 0, 3);
            __builtin_prefetch(Pn + (size_t)48 * D_DIM, 0, 3);
        }

        v8f acc[2][4] = {{v8f{}, v8f{}, v8f{}, v8f{}},
                         {v8f{}, v8f{}, v8f{}, v8f{}}};

        for (int d = 0; d < D_DIM; d += 4) {
            v2f a0 = *(const v2f*)(A0 + d);
            v2f a1 = *(const v2f*)(A1 + d);
            v2f b0 = *(const v2f*)(Bp + d);
            v2f b1 = *(const v2f*)(Bp + (size_t)16 * D_DIM + d);
            v2f b2 = *(const v2f*)(Bp + (size_t)32 * D_DIM + d);
            v2f b3 = *(const v2f*)(Bp + (size_t)48 * D_DIM + d);
            acc[0][0] = __builtin_amdgcn_wmma_f32_16x16x4_f32(
                false, a0, false, b0, (short)0, acc[0][0], false, false);
            acc[0][1] = __builtin_amdgcn_wmma_f32_16x16x4_f32(
                false, a0, false, b1, (short)0, acc[0][1], false, false);
            acc[0][2] = __builtin_amdgcn_wmma_f32_16x16x4_f32(
                false, a0, false, b2, (short)0, acc[0][2], false, false);
            acc[0][3] = __builtin_amdgcn_wmma_f32_16x16x4_f32(
                false, a0, false, b3, (short)0, acc[0][3], false, false);
            acc[1][0] = __builtin_amdgcn_wmma_f32_16x16x4_f32(
                false, a1, false, b0, (short)0, acc[1][0], false, false);
            acc[1][1] = __builtin_amdgcn_wmma_f32_16x16x4_f32(
                false, a1, false, b1, (short)0, acc[1][1], false, false);
            acc[1][2] = __builtin_amdgcn_wmma_f32_16x16x4_f32(
                false, a1, false, b2, (short)0, acc[1][2], false, false);
            acc[1][3] = __builtin_amdgcn_wmma_f32_16x16x4_f32(
                false, a1, false, b3, (short)0, acc[1][3], false, false);
        }

        // Fold this column group into the running min/argmin.
#pragma unroll
        for (int t = 0; t < 4; ++t) {
            const int col = cg + 16 * t + l16;
            const float cc = c2[col];
#pragma unroll
            for (int u = 0; u < 2; ++u) {
#pragma unroll
                for (int j = 0; j < 8; ++j) {
                    float v = __builtin_fmaf(-2.0f, acc[u][t][j], cc);
                    if (v < vmin[u][j]) { vmin[u][j] = v; amin[u][j] = col; }
                }
            }
        }
    }

    // Cross-lane min/argmin over the 16 column-lanes of each half.
#pragma unroll
    for (int off = 1; off < 16; off <<= 1) {
#pragma unroll
        for (int u = 0; u < 2; ++u) {
#pragma unroll
            for (int j = 0; j < 8; ++j) {
                float ov = __shfl_xor(vmin[u][j], off, 32);
                int   oa = __shfl_xor(amin[u][j], off, 32);
                if (ov < vmin[u][j] || (ov == vmin[u][j] && oa < amin[u][j])) {
                    vmin[u][j] = ov; amin[u][j] = oa;
                }
            }
        }
    }

    // Row (within wave tile) r = 16*u + 8*half + j; lane l16==j writes it.
    if (l16 < 8) {
        const int j = l16;
#pragma unroll
        for (int u = 0; u < 2; ++u) {
            const int n = n0 + 16 * u + 8 * half + j;
            const int a = amin[u][j];
            s_out[n] = (float)a;
            s_int[n] = a;
            m_out[n] = vmin[u][j];     // min_k (c2[k] - 2*e.c[k]); e2 added later
        }
    }
}

// ---------------------------------------------------------------------------
// Kernel: loss accumulation, sum_n ( ||e_n||^2 + m[n] ).  One wave per row.
// ---------------------------------------------------------------------------
__global__ __launch_bounds__(256) void loss_kernel(const float* __restrict__ E,
                                                   const float* __restrict__ m,
                                                   float* __restrict__ lossacc) {
    __shared__ float red[8];
    const int lane = threadIdx.x & 31;
    const int w    = threadIdx.x >> 5;
    const int n    = blockIdx.x * 8 + w;
    const float* row = E + (size_t)n * D_DIM;
    float s = 0.0f;
#pragma unroll
    for (int i = 0; i < D_DIM / 128; ++i) {            // b128 per lane
        const float4 v = ((const float4*)row)[lane + 32 * i];
        s += v.x * v.x + v.y * v.y + v.z * v.z + v.w * v.w;
    }
#pragma unroll
    for (int off = 16; off > 0; off >>= 1) s += __shfl_xor(s, off, 32);
    if (lane == 0) red[w] = s + m[n];
    __syncthreads();
    if (threadIdx.x == 0) {
        float t = 0.0f;
#pragma unroll
        for (int j = 0; j < 8; ++j) t += red[j];
        atomicAdd(lossacc, t);
    }
}

// ---------------------------------------------------------------------------
// Kernel: segment sums via global float atomics (one block per sample)
// ---------------------------------------------------------------------------
__global__ __launch_bounds__(128) void scatter_kernel(
    const float* __restrict__ E, const int* __restrict__ s,
    float* __restrict__ added, int* __restrict__ n_add)
{
    const int n = blockIdx.x;
    const int c = s[n];
    const float* e = E + (size_t)n * D_DIM;
    float* a = added + (size_t)c * D_DIM;
    for (int d = threadIdx.x; d < D_DIM; d += 128) atomicAdd(&a[d], e[d]);
    if (threadIdx.x == 0) atomicAdd(&n_add[c], 1);
}

// ---------------------------------------------------------------------------
// Kernel: finalize new_centers, new_count, loss
// ---------------------------------------------------------------------------
__global__ __launch_bounds__(256) void finalize_kernel(
    const float* __restrict__ centers, const int* __restrict__ count,
    const float* __restrict__ added, const int* __restrict__ n_add,
    const float* __restrict__ lossacc, float* __restrict__ out)
{
    const int idx = blockIdx.x * 256 + threadIdx.x;   // [0, K*D)
    const int k = idx >> 9;                            // / D_DIM
    const float cf = (float)count[k];
    const float nc = cf + (float)n_add[k];
    out[1 + N_PTS + idx] = (cf * centers[idx] + added[idx]) / nc;
    if (idx < K_CL) out[1 + N_PTS + K_CL * D_DIM + idx] = (float)(count[idx] + n_add[idx]);
    if (idx == 0) out[0] = lossacc[0] / (float)N_PTS;
}

// ---------------------------------------------------------------------------
// kernel_launch
// d_out layout (reference tuple, flat, as float):
//   [0]                      loss
//   [1 .. 1+N)               s (argmin indices, cast to float)
//   [1+N .. 1+N+K*D)         new_centers
//   [1+N+K*D .. +K)          new_count
// ---------------------------------------------------------------------------
extern "C" void kernel_launch(void* const* d_in, const int* in_sizes, int n_in,
                              void* d_out, int out_size, void* d_ws, size_t ws_size,
                              hipStream_t stream) {
    (void)in_sizes; (void)n_in; (void)out_size; (void)ws_size;
    const float* E      = (const float*)d_in[0];   // [N,D]
    const float* C      = (const float*)d_in[1];   // [K,D]
    const int*   count  = (const int*)d_in[2];     // [K]
    float* out = (float*)d_out;

    // Workspace layout
    float* ws      = (float*)d_ws;
    float* m       = ws;                               // N floats
    int*   s_int   = (int*)(ws + N_PTS);               // N ints
    float* c2      = ws + 2 * N_PTS;                   // K floats
    float* added   = ws + 2 * N_PTS + K_CL;            // K*D floats
    int*   n_add   = (int*)(added + (size_t)K_CL * D_DIM);  // K ints
    float* lossacc = (float*)(n_add + K_CL);           // 1 float

    float* s_out = out + 1;

    // 1) zero added + n_add + lossacc
    const int nz = K_CL * D_DIM + K_CL + 1;
    zero_kernel<<<(nz + 255) / 256, 256, 0, stream>>>(added, nz);

    // 2) c2[k]
    c2_kernel<<<K_CL / 8, 256, 0, stream>>>(C, c2);

    // 3) WMMA distance + argmin (256 blocks x 8 waves x 32 rows = 65536 rows)
    assign_kernel<<<N_PTS / 256, 256, 0, stream>>>(E, C, c2, s_out, m, s_int);

    // 4) loss = sum(e2 + m) -> lossacc
    loss_kernel<<<N_PTS / 8, 256, 0, stream>>>(E, m, lossacc);

    // 5) segment sums
    scatter_kernel<<<N_PTS, 128, 0, stream>>>(E, s_int, added, n_add);

    // 6) finalize centers / counts / loss
    finalize_kernel<<<(K_CL * D_DIM) / 256, 256, 0, stream>>>(
        C, count, added, n_add, lossacc, out);
}